// TreeCRFSimulation_18442589569875
// MI455X (gfx1250) — compile-verified
//
#include <hip/hip_runtime.h>
#include <hip/hip_bf16.h>
#include <math.h>

// Problem constants from the reference
#define Bv 8
#define Lv 96
#define Sv 32
#define Hv 256

typedef __attribute__((ext_vector_type(2))) float v2f;
typedef __attribute__((ext_vector_type(8))) float v8f;

// The entire TreeCRF computation collapses (see analysis) to:
//   s[a]  = sum_h state_matrix[a,h]                     (32x256 row sums -> WMMA)
//   t[a]  = 10*(s[a]-min s)/(max s - min s)
//   T     = LSE_a t[a];  F(0)=T
//   F(d)  = T + LSE_{k<d}( F(k)+F(d-1-k) ),  d=1..95
//   out[b]= F(95) for all b
// word_emb cancels exactly in the normalization and is never read.
__global__ __launch_bounds__(32)
void treecrf_collapse_kernel(const float* __restrict__ state_matrix,
                             float* __restrict__ out) {
  __shared__ float sS[Sv];   // row sums s[a]
  __shared__ float sF[Lv];   // recurrence F(d)

  const int lane = threadIdx.x;          // wave32: 0..31, EXEC all ones

  // ---- Stage 1: s = state_matrix x ones  via V_WMMA_F32_16X16X4_F32 ----
  // A (16x4 f32) lane layout (ISA 7.12.2): lanes 0-15 hold K={0,1},
  // lanes 16-31 hold K={2,3} for rows M = lane&15.
  const int   row  = lane & 15;
  const int   koff = (lane >> 4) << 1;   // 0 or 2
  const float* p0 = state_matrix + row * Hv;          // rows 0..15
  const float* p1 = state_matrix + (16 + row) * Hv;   // rows 16..31

  v8f acc0 = {};   // C/D 16x16 f32: elem r -> M=r (lanes 0-15), M=8+r (lanes 16-31)
  v8f acc1 = {};
  const v2f ones = {1.0f, 1.0f};

#pragma unroll 4
  for (int c = 0; c < Hv / 4; ++c) {
    const int kb = c * 4 + koff;
    v2f a0, a1;
    a0.x = p0[kb];     a0.y = p0[kb + 1];
    a1.x = p1[kb];     a1.y = p1[kb + 1];
    // D = A*ones + C : every column of D holds the (partial) row sum
    acc0 = __builtin_amdgcn_wmma_f32_16x16x4_f32(
        /*neg_a=*/false, a0, /*neg_b=*/false, ones,
        /*c_mod=*/(short)0, acc0, /*reuse_a=*/false, /*reuse_b=*/false);
    acc1 = __builtin_amdgcn_wmma_f32_16x16x4_f32(
        false, a1, false, ones, (short)0, acc1, false, false);
  }

  // Column N=0 lives in lane 0 (rows 0-7 / 16-23) and lane 16 (rows 8-15 / 24-31)
  if (lane == 0) {
#pragma unroll
    for (int r = 0; r < 8; ++r) { sS[r]      = acc0[r]; sS[16 + r] = acc1[r]; }
  }
  if (lane == 16) {
#pragma unroll
    for (int r = 0; r < 8; ++r) { sS[8 + r]  = acc0[r]; sS[24 + r] = acc1[r]; }
  }
  __syncthreads();

  // ---- Stage 2: t[a], T = logsumexp_a t[a] (wave32 shuffle reductions) ----
  float s = sS[lane];
  float smin = s, smax = s;
#pragma unroll
  for (int off = 16; off > 0; off >>= 1) {
    smin = fminf(smin, __shfl_xor(smin, off, 32));
    smax = fmaxf(smax, __shfl_xor(smax, off, 32));
  }
  const float t = 10.0f * (s - smin) / (smax - smin);

  float tmax = t;
#pragma unroll
  for (int off = 16; off > 0; off >>= 1)
    tmax = fmaxf(tmax, __shfl_xor(tmax, off, 32));
  float te = __expf(t - tmax);
#pragma unroll
  for (int off = 16; off > 0; off >>= 1)
    te += __shfl_xor(te, off, 32);
  const float T = tmax + __logf(te);

  if (lane == 0) sF[0] = T;     // F(0) = T
  __syncthreads();

  // ---- Stage 3: F(d) = T + LSE_{k<d}(F(k)+F(d-1-k)), d = 1..95 ----
  for (int d = 1; d < Lv; ++d) {
    float mx = -INFINITY;
    for (int k = lane; k < d; k += 32)
      mx = fmaxf(mx, sF[k] + sF[d - 1 - k]);
#pragma unroll
    for (int off = 16; off > 0; off >>= 1)
      mx = fmaxf(mx, __shfl_xor(mx, off, 32));

    float sum = 0.0f;
    for (int k = lane; k < d; k += 32)
      sum += __expf(sF[k] + sF[d - 1 - k] - mx);
#pragma unroll
    for (int off = 16; off > 0; off >>= 1)
      sum += __shfl_xor(sum, off, 32);

    const float Fd = T + mx + __logf(sum);
    if (lane == 0) sF[d] = Fd;
    __syncthreads();
  }

  // ---- Output: identical for every batch element ----
  if (lane < Bv) out[lane] = sF[Lv - 1];
}

extern "C" void kernel_launch(void* const* d_in, const int* in_sizes, int n_in,
                              void* d_out, int out_size, void* d_ws, size_t ws_size,
                              hipStream_t stream) {
  (void)in_sizes; (void)n_in; (void)d_ws; (void)ws_size; (void)out_size;
  // d_in[0] = word_emb (B,L,H) f32  -- mathematically cancels, never read
  // d_in[1] = state_matrix (S,H) f32
  const float* state = (const float*)d_in[1];
  float* out = (float*)d_out;
  treecrf_collapse_kernel<<<dim3(1), dim3(32), 0, stream>>>(state, out);
}